// LayerNormCompiled_89781996355911
// MI455X (gfx1250) — compile-verified
//
#include <hip/hip_runtime.h>
#include <hip/hip_bf16.h>

// Equivariant norm for irreps 128x0e + 64x1e + 32x2e over [N, 480] f32.
//   cols [0,128):   per-element layer-norm over a size-1 axis -> exactly 0
//                   (numerator f - mean(f) == 0), so no need to even read them.
//   cols [128,320): scale by rsqrt(sumsq/64 + eps)
//   cols [320,480): scale by rsqrt(sumsq/32 + eps)
// Streaming kernel: read 1408 B/row + write 1920 B/row = 436 MB total
// -> ~18.7 us floor at 23.3 TB/s HBM. One wave32 per row, NT b128 access,
// 6-shuffle folded wave reduction.

typedef float v4f __attribute__((ext_vector_type(4)));

#define EQN_EPS   1e-6f
#define N_COLS    480

__global__ __launch_bounds__(256)
void eqnorm_kernel(const float* __restrict__ x, float* __restrict__ out,
                   int nrows, int total_waves) {
    const int lane = threadIdx.x & 31;
    const int wave = (blockIdx.x * (blockDim.x >> 5)) + (threadIdx.x >> 5);

    for (int row = wave; row < nrows; row += total_waves) {
        const v4f* __restrict__ xin = (const v4f*)(x + (size_t)row * N_COLS);
        v4f* __restrict__ xout      = (v4f*)(out + (size_t)row * N_COLS);

        // Speculative prefetch of the next row's *read* region (bytes [512,1920)):
        // 32 lanes x 44B stride covers every cacheline (global_prefetch_b8).
        if (row + total_waves < nrows) {
            __builtin_prefetch(x + (size_t)(row + total_waves) * N_COLS + 128 + lane * 11, 0, 0);
        }

        // Lane L owns float4 chunks L+32, L+64, L+96(<120) of the 120-chunk row.
        // chunk c covers elements [4c, 4c+4):
        //   c in [0,32)   -> scalar segment: output 0, never loaded
        //   c in [32,80)  -> 1e segment
        //   c in [80,120) -> 2e segment
        v4f v1 = __builtin_nontemporal_load(&xin[lane + 32]);   // 1e
        v4f v2 = __builtin_nontemporal_load(&xin[lane + 64]);   // 1e (lane<16) / 2e
        v4f v3 = {0.0f, 0.0f, 0.0f, 0.0f};
        const bool has3 = (lane < 24);                          // chunks 96..119
        if (has3) v3 = __builtin_nontemporal_load(&xin[lane + 96]);

        float ss1 = v1.x * v1.x + v1.y * v1.y + v1.z * v1.z + v1.w * v1.w;
        float ss2 = v3.x * v3.x + v3.y * v3.y + v3.z * v3.z + v3.w * v3.w;
        float mid = v2.x * v2.x + v2.y * v2.y + v2.z * v2.z + v2.w * v2.w;
        const bool lo = (lane < 16);
        if (lo) ss1 += mid; else ss2 += mid;

        // Folded wave32 reduction: lanes 0-15 reduce ss1, lanes 16-31 reduce ss2.
        // 6 lane-permutes total instead of 10.
        float t = lo ? ss1 : ss2;
        float u = lo ? ss2 : ss1;
        t += __shfl_xor(u, 16, 32);        // merge cross partials into own half
        #pragma unroll
        for (int m = 8; m >= 1; m >>= 1)   // butterfly within each 16-lane half
            t += __shfl_xor(t, m, 32);
        const float o = __shfl_xor(t, 16, 32);  // fetch the other half's total
        const float S1 = lo ? t : o;
        const float S2 = lo ? o : t;

        const float inv1 = __builtin_amdgcn_rsqf(S1 * (1.0f / 64.0f) + EQN_EPS);
        const float inv2 = __builtin_amdgcn_rsqf(S2 * (1.0f / 32.0f) + EQN_EPS);

        const v4f zero = {0.0f, 0.0f, 0.0f, 0.0f};
        __builtin_nontemporal_store(zero, &xout[lane]);              // scalars -> 0
        __builtin_nontemporal_store(v1 * inv1, &xout[lane + 32]);    // 1e
        const float invm = lo ? inv1 : inv2;
        __builtin_nontemporal_store(v2 * invm, &xout[lane + 64]);    // 1e/2e split
        if (has3) __builtin_nontemporal_store(v3 * inv2, &xout[lane + 96]); // 2e
    }
}

extern "C" void kernel_launch(void* const* d_in, const int* in_sizes, int n_in,
                              void* d_out, int out_size, void* d_ws, size_t ws_size,
                              hipStream_t stream) {
    (void)n_in; (void)d_ws; (void)ws_size; (void)out_size;

    const float* x = (const float*)d_in[0];
    float* out     = (float*)d_out;

    const int nrows = in_sizes[0] / N_COLS;   // 131072 for the reference shapes
    if (nrows <= 0) return;

    const int threads = 256;                  // 8 wave32 per block
    const int waves_per_block = threads / 32;
    int blocks = (nrows + waves_per_block - 1) / waves_per_block;
    if (blocks > 4096) blocks = 4096;         // grid-stride: ~4 rows per wave
    const int total_waves = blocks * waves_per_block;

    eqnorm_kernel<<<blocks, threads, 0, stream>>>(x, out, nrows, total_waves);
}